// Sinkhorn_57801669870081
// MI455X (gfx1250) — compile-verified
//
#include <hip/hip_runtime.h>
#include <math.h>

#define BDIM 2048
#define CDIM 1024
#define LOG2E 1.4426950408889634f
#define KFAC  (-14.426950408889634f)   /* -10 * log2(e):  exp(-W/0.1) = exp2(W*KFAC) */

typedef float v2f __attribute__((ext_vector_type(2)));
typedef float v8f __attribute__((ext_vector_type(8)));

__device__ __forceinline__ float fast_exp2(float x) {
#if __has_builtin(__builtin_amdgcn_exp2f)
  return __builtin_amdgcn_exp2f(x);
#else
  return exp2f(x);
#endif
}

__device__ __forceinline__ float block_reduce_sum256(float x) {
  __shared__ float sm[8];
  const int lane = threadIdx.x & 31;
  const int wid  = threadIdx.x >> 5;
#pragma unroll
  for (int o = 16; o > 0; o >>= 1) x += __shfl_xor(x, o, 32);
  if (lane == 0) sm[wid] = x;
  __syncthreads();
  float r = 0.f;
#pragma unroll
  for (int k = 0; k < 8; ++k) r += sm[k];
  __syncthreads();
  return r;
}

__device__ __forceinline__ float block_reduce_max256(float x) {
  __shared__ float sm[8];
  const int lane = threadIdx.x & 31;
  const int wid  = threadIdx.x >> 5;
#pragma unroll
  for (int o = 16; o > 0; o >>= 1) x = fmaxf(x, __shfl_xor(x, o, 32));
  if (lane == 0) sm[wid] = x;
  __syncthreads();
  float r = sm[0];
#pragma unroll
  for (int k = 1; k < 8; ++k) r = fmaxf(r, sm[k]);
  __syncthreads();
  return r;
}

/* ---------------- softmax(y/2) per row, computed in exp2 domain ---------------- */
__global__ __launch_bounds__(256)
void softmax_rows(const float* __restrict__ y, float* __restrict__ p) {
  const int row = blockIdx.x;
  const float* yr = y + (size_t)row * CDIM;
  float* pr = p + (size_t)row * CDIM;
  float v[4];
  float mx = -3.0e38f;
#pragma unroll
  for (int q = 0; q < 4; ++q) {
    v[q] = yr[threadIdx.x + q * 256] * (0.5f * LOG2E);
    mx = fmaxf(mx, v[q]);
  }
  mx = block_reduce_max256(mx);
  float s = 0.f;
#pragma unroll
  for (int q = 0; q < 4; ++q) { v[q] = fast_exp2(v[q] - mx); s += v[q]; }
  s = block_reduce_sum256(s);
  const float inv = 1.0f / s;
#pragma unroll
  for (int q = 0; q < 4; ++q) pr[threadIdx.x + q * 256] = v[q] * inv;
}

/* ---------------- W[i,j] = sum_c |p_s[i,c] - p_t[j,c]|  (128x128 tiles) -------- */
#define TI 128
#define CK 16
#define LDP (TI + 8)   /* 136 floats: 16B-aligned rows, spread banks */

__global__ __launch_bounds__(256)
void cdist_l1(const float* __restrict__ ps, const float* __restrict__ pt,
              float* __restrict__ W) {
  __shared__ float As[CK][LDP];   /* transposed: As[c][row] */
  __shared__ float Bs[CK][LDP];
  const int t  = threadIdx.x;
  const int tx = t & 15;
  const int ty = t >> 4;
  const int i0 = blockIdx.y * TI;
  const int j0 = blockIdx.x * TI;

  float acc[8][8];
#pragma unroll
  for (int i = 0; i < 8; ++i)
#pragma unroll
    for (int j = 0; j < 8; ++j) acc[i][j] = 0.f;

  for (int cb = 0; cb < CDIM; cb += CK) {
#pragma unroll
    for (int q = 0; q < 2; ++q) {
      const int f   = t + q * 256;        /* 0..511 float4 slots per matrix */
      const int row = f >> 2;             /* 4 float4 per 16-col row */
      const int c4  = (f & 3) << 2;
      const float4 a = *(const float4*)(ps + (size_t)(i0 + row) * CDIM + cb + c4);
      const float4 b = *(const float4*)(pt + (size_t)(j0 + row) * CDIM + cb + c4);
      As[c4 + 0][row] = a.x; As[c4 + 1][row] = a.y;
      As[c4 + 2][row] = a.z; As[c4 + 3][row] = a.w;
      Bs[c4 + 0][row] = b.x; Bs[c4 + 1][row] = b.y;
      Bs[c4 + 2][row] = b.z; Bs[c4 + 3][row] = b.w;
    }
    __syncthreads();

    if (cb + CK < CDIM) {   /* CDNA5 global_prefetch_b8 for next chunk */
      const int row = t >> 2;
      const int c4  = (t & 3) << 2;
      __builtin_prefetch(ps + (size_t)(i0 + row) * CDIM + cb + CK + c4, 0, 1);
      __builtin_prefetch(pt + (size_t)(j0 + row) * CDIM + cb + CK + c4, 0, 1);
    }

#pragma unroll
    for (int c = 0; c < CK; ++c) {
      const float4 a0 = *(const float4*)&As[c][ty * 8];
      const float4 a1 = *(const float4*)&As[c][ty * 8 + 4];
      const float4 b0 = *(const float4*)&Bs[c][tx * 8];
      const float4 b1 = *(const float4*)&Bs[c][tx * 8 + 4];
      const float aa[8] = {a0.x, a0.y, a0.z, a0.w, a1.x, a1.y, a1.z, a1.w};
      const float bb[8] = {b0.x, b0.y, b0.z, b0.w, b1.x, b1.y, b1.z, b1.w};
#pragma unroll
      for (int i = 0; i < 8; ++i)
#pragma unroll
        for (int j = 0; j < 8; ++j)
          acc[i][j] += fabsf(aa[i] - bb[j]);
    }
    __syncthreads();
  }

#pragma unroll
  for (int i = 0; i < 8; ++i) {
    float* wr = W + (size_t)(i0 + ty * 8 + i) * BDIM + j0 + tx * 8;
    const float4 o0 = {acc[i][0], acc[i][1], acc[i][2], acc[i][3]};
    const float4 o1 = {acc[i][4], acc[i][5], acc[i][6], acc[i][7]};
    *(float4*)(wr)     = o0;
    *(float4*)(wr + 4) = o1;
  }
}

/* ---------------- Sinkhorn matvec via V_WMMA_F32_16X16X4_F32 -------------------
 * out = 1 / (K   . x)   when TRANS==0   (K_ij = exp2(KFAC * W_ij))
 * out = 1 / (K^T . x)   when TRANS==1
 * Block: 256 threads = 8 waves; each wave owns one 256-wide K-slice of the same
 * 16 output rows; partials combined through LDS.
 * Layouts (ISA 7.12.2, 32-bit): A lane L -> M = L&15, K = ((L>>4)<<1)+{0,1};
 * B VGPR0 = rows K=0,2 / VGPR1 = rows K=1,3 (broadcast x across all 16 N so
 * every D column equals the matvec result); D col N=0 -> lanes 0 (M=0..7) and
 * 16 (M=8..15).
 * ----------------------------------------------------------------------------- */
#if __has_builtin(__builtin_amdgcn_wmma_f32_16x16x4_f32)
#define HAVE_WMMA_F32X4 1
#else
#define HAVE_WMMA_F32X4 0
#endif

template <int TRANS>
__global__ __launch_bounds__(256)
void sinkhorn_matvec(const float* __restrict__ W, const float* __restrict__ x,
                     float* __restrict__ out) {
  __shared__ float red[8][16];
  const int t    = threadIdx.x;
  const int lane = t & 31;
  const int wid  = t >> 5;
  const int base = blockIdx.x * 16;          /* 16 output rows per block */
  const int m    = lane & 15;
  const int k0   = (lane >> 4) << 1;         /* 0 or 2 */
  const int kbeg = wid * (BDIM / 8);
  const int kend = kbeg + (BDIM / 8);

#if HAVE_WMMA_F32X4
  v8f acc = {0.f, 0.f, 0.f, 0.f, 0.f, 0.f, 0.f, 0.f};
  for (int kb = kbeg; kb < kend; kb += 4) {
    float w0, w1;
    if (TRANS == 0) {
      const float* p = W + (size_t)(base + m) * BDIM + kb + k0;
      w0 = p[0]; w1 = p[1];
    } else {
      w0 = W[(size_t)(kb + k0)     * BDIM + base + m];
      w1 = W[(size_t)(kb + k0 + 1) * BDIM + base + m];
    }
    v2f a, b;
    a.x = fast_exp2(w0 * KFAC);
    a.y = fast_exp2(w1 * KFAC);
    b.x = x[kb + k0];
    b.y = x[kb + k0 + 1];
    acc = __builtin_amdgcn_wmma_f32_16x16x4_f32(false, a, false, b,
                                                (short)0, acc, false, false);
  }
  if ((lane & 15) == 0) {
    const int ro = (lane >> 4) << 3;
#pragma unroll
    for (int p = 0; p < 8; ++p) red[wid][ro + p] = acc[p];
  }
#else
  /* scalar fallback: halves of the wave split the k range for row m */
  float s = 0.f;
  for (int kb = kbeg + (lane >> 4); kb < kend; kb += 2) {
    const float w = (TRANS == 0) ? W[(size_t)(base + m) * BDIM + kb]
                                 : W[(size_t)kb * BDIM + base + m];
    s += fast_exp2(w * KFAC) * x[kb];
  }
  s += __shfl_xor(s, 16, 32);
  if (lane < 16) red[wid][m] = s;
#endif
  __syncthreads();
  if (t < 16) {
    float s = 0.f;
#pragma unroll
    for (int w = 0; w < 8; ++w) s += red[w][t];
    out[base + t] = 1.0f / s;
  }
}

/* ---------------- loss = 0.001 * sum u_i K_ij v_j W_ij (deterministic) --------- */
__global__ __launch_bounds__(256)
void loss_rows(const float* __restrict__ W, const float* __restrict__ u,
               const float* __restrict__ v, float* __restrict__ partial) {
  const int i = blockIdx.x;
  const float* wr = W + (size_t)i * BDIM;
  float s = 0.f;
  for (int j = threadIdx.x; j < BDIM; j += 256) {
    const float w = wr[j];
    s += fast_exp2(w * KFAC) * v[j] * w;
  }
  s = block_reduce_sum256(s);
  if (threadIdx.x == 0) partial[i] = s * u[i];
}

__global__ __launch_bounds__(256)
void loss_final(const float* __restrict__ partial, float* __restrict__ out) {
  float s = 0.f;
  for (int j = threadIdx.x; j < BDIM; j += 256) s += partial[j];
  s = block_reduce_sum256(s);
  if (threadIdx.x == 0) out[0] = 0.001f * s;
}

__global__ void fill_ones(float* __restrict__ p, int n) {
  const int i = blockIdx.x * blockDim.x + threadIdx.x;
  if (i < n) p[i] = 1.0f;
}

extern "C" void kernel_launch(void* const* d_in, const int* in_sizes, int n_in,
                              void* d_out, int out_size, void* d_ws, size_t ws_size,
                              hipStream_t stream) {
  (void)in_sizes; (void)n_in; (void)out_size; (void)ws_size;
  const float* y_s = (const float*)d_in[0];
  const float* y_t = (const float*)d_in[1];

  float* ws = (float*)d_ws;
  float* ps      = ws;                               /* 2048*1024 */
  float* pt      = ps + (size_t)BDIM * CDIM;         /* 2048*1024 */
  float* W       = pt + (size_t)BDIM * CDIM;         /* 2048*2048 */
  float* u       = W  + (size_t)BDIM * BDIM;         /* 2048 */
  float* v       = u + BDIM;                         /* 2048 */
  float* partial = v + BDIM;                         /* 2048 */

  softmax_rows<<<BDIM, 256, 0, stream>>>(y_s, ps);
  softmax_rows<<<BDIM, 256, 0, stream>>>(y_t, pt);

  cdist_l1<<<dim3(BDIM / TI, BDIM / TI), 256, 0, stream>>>(ps, pt, W);

  fill_ones<<<BDIM / 256, 256, 0, stream>>>(v, BDIM);

  for (int it = 0; it < 20; ++it) {
    sinkhorn_matvec<0><<<BDIM / 16, 256, 0, stream>>>(W, v, u);  /* u = 1/(K v)   */
    sinkhorn_matvec<1><<<BDIM / 16, 256, 0, stream>>>(W, u, v);  /* v = 1/(K^T u) */
  }

  loss_rows<<<BDIM, 256, 0, stream>>>(W, u, v, partial);
  loss_final<<<1, 256, 0, stream>>>(partial, (float*)d_out);
}